// CBAMAttention_70480413327661
// MI455X (gfx1250) — compile-verified
//
#include <hip/hip_runtime.h>
#include <math.h>

// CBAM: x[32,256,64,64] f32. Memory-bound (~512 MB traffic; x fits in 192MB L2
// so passes 2/3 re-read from L2). MLP done with V_WMMA_F32_16X16X4_F32.
// Final pass uses non-temporal loads/stores so the 128MB output stream does
// not evict x/sig from L2 mid-pass.

#define B_   32
#define C_   256
#define H_   64
#define W_   64
#define HW_  4096
#define MID_ 16

typedef __attribute__((ext_vector_type(2))) float v2f;
typedef __attribute__((ext_vector_type(4))) float v4f;
typedef __attribute__((ext_vector_type(8))) float v8f;

// ---------------- Kernel 1: global avg+max pool over H*W per (b,c) ----------
__global__ __launch_bounds__(256) void cbam_pool(const float* __restrict__ x,
                                                 float* __restrict__ avg,
                                                 float* __restrict__ mxp) {
  int plane = blockIdx.x;                       // b*C_ + c; plane = 4096 contiguous floats
  const v4f* p = (const v4f*)(x + (size_t)plane * HW_);
  float s = 0.f, m = -INFINITY;
#pragma unroll
  for (int i = 0; i < 4; ++i) {
    v4f v = p[threadIdx.x + i * 256];
    s += v.x + v.y + v.z + v.w;
    m = fmaxf(m, fmaxf(fmaxf(v.x, v.y), fmaxf(v.z, v.w)));
  }
  // wave32 reduction
#pragma unroll
  for (int off = 16; off >= 1; off >>= 1) {
    s += __shfl_xor(s, off, 32);
    m = fmaxf(m, __shfl_xor(m, off, 32));
  }
  __shared__ float ss[8], sm[8];
  int wave = threadIdx.x >> 5, lane = threadIdx.x & 31;
  if (lane == 0) { ss[wave] = s; sm[wave] = m; }
  __syncthreads();
  if (threadIdx.x == 0) {
    float ts = 0.f, tm = -INFINITY;
#pragma unroll
    for (int i = 0; i < 8; ++i) { ts += ss[i]; tm = fmaxf(tm, sm[i]); }
    avg[plane] = ts * (1.0f / (float)HW_);
    mxp[plane] = tm;
  }
}

// ---------------- Kernel 2: shared MLP via WMMA f32 16x16x4 + sigmoid -------
// h_s = relu(pool_s[32,256] @ w1^T[256,16])   (s in {avg,max})
// ca  = sigmoid([h_avg|h_max][32,32] @ [w2^T;w2^T][32,256])
// One block, 8 waves. Phase 1: 4 waves each own one (src, mtile) 16x16 tile.
// Phase 2: 32 output tiles, 4 per wave.
__global__ __launch_bounds__(256) void cbam_mlp_ca(const float* __restrict__ avg,
                                                   const float* __restrict__ mxp,
                                                   const float* __restrict__ w1,
                                                   const float* __restrict__ w2,
                                                   float* __restrict__ ca) {
  __shared__ float hcomb[32][32];   // [batch 0..31][k: 0..15 = h_avg, 16..31 = h_max]
  int tid  = threadIdx.x;
  int wave = tid >> 5;
  int lane = tid & 31;
  int half = lane >> 4;             // lane half-group (ISA f32 operand layout)
  int l15  = lane & 15;

  if (wave < 4) {
    int src   = wave >> 1;          // 0 = avg path, 1 = max path
    int mtile = wave & 1;           // batch rows [mtile*16, mtile*16+16)
    const float* pool = src ? mxp : avg;
    int M = mtile * 16 + l15;       // A-matrix: lane holds row M = lane%16
    v8f acc = {};
    for (int k0 = 0; k0 < 256; k0 += 4) {
      int ka = k0 + half * 2;       // A/B VGPR0 holds K=ka, VGPR1 holds K=ka+1
      v2f a, b;
      a.x = pool[M * 256 + ka];
      a.y = pool[M * 256 + ka + 1];
      b.x = w1[l15 * 256 + ka];     // B[k][n] = w1[n][k], n = lane%16
      b.y = w1[l15 * 256 + ka + 1];
      acc = __builtin_amdgcn_wmma_f32_16x16x4_f32(
          false, a, false, b, (short)0, acc, false, false);
    }
#pragma unroll
    for (int i = 0; i < 8; ++i) {   // D: VGPR i -> M = i + half*8, N = lane%16
      int m = mtile * 16 + i + half * 8;
      hcomb[m][src * 16 + l15] = fmaxf(acc[i], 0.f);   // relu
    }
  }
  __syncthreads();

  for (int t = wave; t < 32; t += 8) {
    int mtile = t >> 4;             // 0..1 (batch tile)
    int ntile = t & 15;             // 0..15 (channel tile)
    v8f acc = {};
#pragma unroll
    for (int k0 = 0; k0 < 32; k0 += 4) {
      int ka = k0 + half * 2;
      v2f a, b;
      a.x = hcomb[mtile * 16 + l15][ka];
      a.y = hcomb[mtile * 16 + l15][ka + 1];
      // stacked W2^T: B[k][c] = w2[c][k % 16]
      b.x = w2[(ntile * 16 + l15) * 16 + (ka & 15)];
      b.y = w2[(ntile * 16 + l15) * 16 + ((ka + 1) & 15)];
      acc = __builtin_amdgcn_wmma_f32_16x16x4_f32(
          false, a, false, b, (short)0, acc, false, false);
    }
#pragma unroll
    for (int i = 0; i < 8; ++i) {
      int m = mtile * 16 + i + half * 8;
      int c = ntile * 16 + l15;
      ca[m * 256 + c] = 1.f / (1.f + __expf(-acc[i]));  // sigmoid
    }
  }
}

// ---------------- Kernel 3: channel mean/max of (x * ca) --------------------
__global__ __launch_bounds__(256) void cbam_feat(const float* __restrict__ x,
                                                 const float* __restrict__ ca,
                                                 float* __restrict__ fm,
                                                 float* __restrict__ fx) {
  __shared__ float sCa[256];
  int b  = blockIdx.x >> 4;
  int hw = ((blockIdx.x & 15) << 8) + threadIdx.x;
  sCa[threadIdx.x] = ca[b * 256 + threadIdx.x];
  __syncthreads();
  const float* xb = x + (size_t)b * C_ * HW_;
  float s = 0.f, m = -INFINITY;
#pragma unroll 4
  for (int c = 0; c < 256; ++c) {            // coalesced: adjacent threads adjacent hw
    float v = xb[(size_t)c * HW_ + hw] * sCa[c];
    s += v;
    m = fmaxf(m, v);
  }
  fm[b * HW_ + hw] = s * (1.f / 256.f);
  fx[b * HW_ + hw] = m;
}

// ---------------- Kernel 4: 7x7 conv on [mean,max] + sigmoid ----------------
__global__ __launch_bounds__(256) void cbam_conv(const float* __restrict__ fm,
                                                 const float* __restrict__ fx,
                                                 const float* __restrict__ wsp,
                                                 float* __restrict__ sig) {
  __shared__ float sW[98];
  if (threadIdx.x < 98) sW[threadIdx.x] = wsp[threadIdx.x];
  __syncthreads();
  int b  = blockIdx.x >> 4;
  int hw = ((blockIdx.x & 15) << 8) + threadIdx.x;
  int y = hw >> 6, xc = hw & 63;
  const float* fmb = fm + b * HW_;
  const float* fxb = fx + b * HW_;
  float acc = 0.f;
#pragma unroll
  for (int ky = 0; ky < 7; ++ky) {
    int yy = y + ky - 3;
    if (yy < 0 || yy >= H_) continue;
#pragma unroll
    for (int kx = 0; kx < 7; ++kx) {
      int x2 = xc + kx - 3;
      if (x2 < 0 || x2 >= W_) continue;
      acc += fmb[yy * 64 + x2] * sW[ky * 7 + kx]
           + fxb[yy * 64 + x2] * sW[49 + ky * 7 + kx];
    }
  }
  sig[b * HW_ + hw] = 1.f / (1.f + __expf(-acc));
}

// ---------------- Kernel 5: out = x * ca[b,c] * sig[b,hw] -------------------
// Non-temporal on x (last use) and out (write-once) to preserve L2 residency.
__global__ __launch_bounds__(256) void cbam_final(const float* __restrict__ x,
                                                  const float* __restrict__ ca,
                                                  const float* __restrict__ sig,
                                                  float* __restrict__ out) {
  size_t idx = (size_t)blockIdx.x * 256 + threadIdx.x;  // v4f index
  size_t v   = idx * 4;
  int hw = (int)(v & 4095);
  int c  = (int)((v >> 12) & 255);
  int b  = (int)(v >> 20);
  v4f xv = __builtin_nontemporal_load((const v4f*)x + idx);
  v4f sv = *(const v4f*)(sig + (size_t)b * HW_ + hw);
  float cv = ca[b * 256 + c];
  v4f o = xv * (sv * cv);
  __builtin_nontemporal_store(o, (v4f*)out + idx);
}

// ---------------- launch ----------------------------------------------------
extern "C" void kernel_launch(void* const* d_in, const int* in_sizes, int n_in,
                              void* d_out, int out_size, void* d_ws, size_t ws_size,
                              hipStream_t stream) {
  const float* x  = (const float*)d_in[0];   // [32,256,64,64]
  const float* w1 = (const float*)d_in[1];   // [16,256]
  const float* w2 = (const float*)d_in[2];   // [256,16]
  const float* ws = (const float*)d_in[3];   // [1,2,7,7]
  float* out = (float*)d_out;

  float* wsf = (float*)d_ws;
  float* avg = wsf;                          // 8192
  float* mxp = wsf + 8192;                   // 8192
  float* ca  = wsf + 16384;                  // 8192
  float* fm  = wsf + 24576;                  // 131072
  float* fx  = wsf + 155648;                 // 131072
  float* sig = wsf + 286720;                 // 131072   (total ~1.67 MB)

  cbam_pool  <<<B_ * C_,          256, 0, stream>>>(x, avg, mxp);
  cbam_mlp_ca<<<1,                256, 0, stream>>>(avg, mxp, w1, w2, ca);
  cbam_feat  <<<B_ * (HW_ / 256), 256, 0, stream>>>(x, ca, fm, fx);
  cbam_conv  <<<B_ * (HW_ / 256), 256, 0, stream>>>(fm, fx, ws, sig);
  cbam_final <<<(B_ * C_ * HW_ / 4) / 256, 256, 0, stream>>>(x, ca, sig, out);
}